// GCN_36086315221436
// MI455X (gfx1250) — compile-verified
//
#include <hip/hip_runtime.h>

typedef __attribute__((ext_vector_type(2))) float v2f;
typedef __attribute__((ext_vector_type(8))) float v8f;

#define GCN_N_NODES 50000
#define GCN_D_IN    256
#define GCN_D_OUT   128

// ---------------------------------------------------------------- utilities
__global__ void gcn_zero_kernel(float* __restrict__ p, int n) {
    int i = blockIdx.x * blockDim.x + threadIdx.x;
    if (i < n) p[i] = 0.0f;
}

// deg[dst[e]] += 1
__global__ void gcn_degree_kernel(const int* __restrict__ dst,
                                  float* __restrict__ deg, int nedges) {
    int i = blockIdx.x * blockDim.x + threadIdx.x;
    if (i < nedges) atomicAdd(&deg[dst[i]], 1.0f);
}

// deg -> rsqrt(max(deg,1)) in place
__global__ void gcn_norm_kernel(float* __restrict__ deg, int n) {
    int i = blockIdx.x * blockDim.x + threadIdx.x;
    if (i < n) deg[i] = 1.0f / sqrtf(fmaxf(deg[i], 1.0f));
}

// ------------------------------------------------- WMMA GEMM: hlin = (h*norm) @ W^T
// One wave computes a 16-node x 128-output stripe.
// A tile (16x4 f32): lane<16 -> {M=lane, K=k0..k0+1}; lane>=16 -> {M=lane-16, K=k0+2..k0+3}
// B tile (4x16 f32): same split over K, N = lane&15  (B[k][o] = W[o][k], W row-major 256)
// D tile (16x16 f32, 8 VGPRs): VGPR v -> M=v (lanes 0-15) / M=v+8 (lanes 16-31), N=lane&15
__global__ void __launch_bounds__(256)
gcn_gemm_wmma(const float* __restrict__ h, const float* __restrict__ Wm,
              const float* __restrict__ norm, float* __restrict__ hlin) {
    int wave = (int)((blockIdx.x * blockDim.x + threadIdx.x) >> 5);
    int lane = threadIdx.x & 31;
    if (wave >= GCN_N_NODES / 16) return;   // uniform across the wave (3125 tiles exactly)

    const int m_base = wave << 4;
    const int row    = lane & 15;
    const int koff   = (lane >> 4) << 1;    // 0 or 2

    const float* hrow  = h  + (size_t)(m_base + row) * GCN_D_IN + koff;
    const float* wbase = Wm + (size_t)row * GCN_D_IN + koff;
    const float  nrm   = norm[m_base + row];

    v8f acc[8];
#pragma unroll
    for (int t = 0; t < 8; ++t)
        acc[t] = (v8f){0.f, 0.f, 0.f, 0.f, 0.f, 0.f, 0.f, 0.f};

    for (int k0 = 0; k0 < GCN_D_IN; k0 += 4) {
        v2f a = *(const v2f*)(hrow + k0);
        a *= nrm;                            // fold pre-normalization into A
#pragma unroll
        for (int t = 0; t < 8; ++t) {
            v2f b = *(const v2f*)(wbase + (size_t)t * 16 * GCN_D_IN + k0);
            acc[t] = __builtin_amdgcn_wmma_f32_16x16x4_f32(
                false, a, false, b, (short)0, acc[t], false, false);
        }
    }

    const int hi  = (lane >> 4) << 3;        // 0 or 8
    const int col = lane & 15;
#pragma unroll
    for (int t = 0; t < 8; ++t) {
#pragma unroll
        for (int v = 0; v < 8; ++v) {
            hlin[(size_t)(m_base + hi + v) * GCN_D_OUT + t * 16 + col] = acc[t][v];
        }
    }
}

// ------------------------------------------------- edge scatter: out[dst] += hlin[src]
// One wave per edge; each lane owns 4 consecutive features (float4).
__global__ void gcn_scatter_kernel(const float* __restrict__ hlin,
                                   const int* __restrict__ src,
                                   const int* __restrict__ dst,
                                   float* __restrict__ out, int nedges) {
    int gid  = blockIdx.x * blockDim.x + threadIdx.x;
    int wave = gid >> 5;
    int lane = threadIdx.x & 31;
    if (wave >= nedges) return;
    int s = src[wave];
    int d = dst[wave];
    const float4 v = *(const float4*)(hlin + (size_t)s * GCN_D_OUT + (lane << 2));
    float* o = out + (size_t)d * GCN_D_OUT + (lane << 2);
    atomicAdd(o + 0, v.x);
    atomicAdd(o + 1, v.y);
    atomicAdd(o + 2, v.z);
    atomicAdd(o + 3, v.w);
}

// ------------------------------------------------- post-normalize: out[n,:] *= norm[n]
__global__ void gcn_scale_kernel(float* __restrict__ out,
                                 const float* __restrict__ norm, int total) {
    int i = blockIdx.x * blockDim.x + threadIdx.x;
    if (i < total) out[i] *= norm[i >> 7];   // D_OUT == 128
}

// ---------------------------------------------------------------- launcher
extern "C" void kernel_launch(void* const* d_in, const int* in_sizes, int n_in,
                              void* d_out, int out_size, void* d_ws, size_t ws_size,
                              hipStream_t stream) {
    const float* h   = (const float*)d_in[0];
    const float* Wm  = (const float*)d_in[1];
    const int*   src = (const int*)d_in[2];
    const int*   dst = (const int*)d_in[3];
    float*       out = (float*)d_out;
    const int nedges = in_sizes[2];

    // workspace layout: [ hlin: 50000*128 f32 | norm: 50000 f32 ]
    float* hlin = (float*)d_ws;
    float* norm = hlin + (size_t)GCN_N_NODES * GCN_D_OUT;

    const int total_out = GCN_N_NODES * GCN_D_OUT;   // 6,400,000

    // 1) zero accumulators (d_out is accumulated via atomics; must be zeroed each call)
    gcn_zero_kernel<<<(total_out + 255) / 256, 256, 0, stream>>>(out, total_out);
    gcn_zero_kernel<<<(GCN_N_NODES + 255) / 256, 256, 0, stream>>>(norm, GCN_N_NODES);

    // 2) in-degree then norm = rsqrt(max(deg,1))
    gcn_degree_kernel<<<(nedges + 255) / 256, 256, 0, stream>>>(dst, norm, nedges);
    gcn_norm_kernel<<<(GCN_N_NODES + 255) / 256, 256, 0, stream>>>(norm, GCN_N_NODES);

    // 3) WMMA GEMM: hlin = (h * norm) @ W^T    (3125 waves, 8 waves/block)
    const int ntiles  = GCN_N_NODES / 16;            // 3125
    const int gblocks = (ntiles + 7) / 8;
    gcn_gemm_wmma<<<gblocks, 256, 0, stream>>>(h, Wm, norm, hlin);

    // 4) scatter-add over edges (one wave per edge, L2-resident working set)
    const int sblocks = (nedges + 7) / 8;            // 8 waves per 256-thread block
    gcn_scatter_kernel<<<sblocks, 256, 0, stream>>>(hlin, src, dst, out, nedges);

    // 5) post-normalization
    gcn_scale_kernel<<<(total_out + 255) / 256, 256, 0, stream>>>(out, norm, total_out);
}